// ChebConv2D_17497696764268
// MI455X (gfx1250) — compile-verified
//
#include <hip/hip_runtime.h>
#include <hip/hip_bf16.h>

// ChebConv2D on MI455X (gfx1250).
// B=8, N=200, C=32, OUT=64, K1=K2=3.
// All heavy math runs through V_WMMA_F32_16X16X4_F32 (fp32 in, fp32 acc),
// preserving the reference's fp32 precision. Working set per batch
// (~51 MB) is L2-resident on MI455X (192 MB L2), so tiles load operands
// straight from global (L2 hits) and we skip LDS staging.
//
// K = 200 is an exact multiple of 4, so K-loops need no edge predication.
// L is padded to 208 rows only so A-fragment row loads (m up to 207) stay
// in-bounds; those rows feed output rows >= 200 which are never stored.
//
// Workspace layout (floats):
//   Lp  : 208*208 padded L
//   LpT : 208*208 padded L^T
//   F   : 9 * 200*200*32 feature slices (Chebyshev features, per batch)
// Total ~44.3 MB.

typedef __attribute__((ext_vector_type(2))) float v2f;
typedef __attribute__((ext_vector_type(8))) float v8f;

#define NPAD 208
#define NN   200
#define CCH  32
#define NCOL 6400            // 200*32
#define SLICE 1280000        // 200*200*32

__device__ __forceinline__ v8f wmma4(v2f a, v2f b, v8f c) {
  // D = A(16x4,f32) * B(4x16,f32) + C(16x16,f32)
  return __builtin_amdgcn_wmma_f32_16x16x4_f32(
      /*neg_a=*/false, a, /*neg_b=*/false, b,
      /*c_mod=*/(short)0, c, /*reuse_a=*/false, /*reuse_b=*/false);
}

// ---------------------------------------------------------------------------
// Normalized adjacency: L[i,j] = (i!=j) * csum(i)^-1/2 * adj[i,j] * rsum(j)^-1/2
// Stored padded 208x208 (zeros outside 200x200), plus transpose.
// ---------------------------------------------------------------------------
__global__ void lap_kernel(const float* __restrict__ adj,
                           float* __restrict__ Lp,
                           float* __restrict__ LpT) {
  __shared__ float cs[256];
  __shared__ float rs[256];
  int t = threadIdx.x;
  if (t < NN) {
    float c = 0.f, r = 0.f;
    for (int i = 0; i < NN; ++i) {
      if (i != t) {
        c += adj[i * NN + t];   // column sum of column t
        r += adj[t * NN + i];   // row sum of row t
      }
    }
    cs[t] = (c > 0.f) ? rsqrtf(c) : 0.f;  // inf -> 0 guard
    rs[t] = (r > 0.f) ? rsqrtf(r) : 0.f;
  }
  __syncthreads();
  for (int idx = t; idx < NPAD * NPAD; idx += blockDim.x) {
    int i = idx / NPAD, j = idx % NPAD;
    float v = 0.f;
    if (i < NN && j < NN && i != j) v = cs[i] * adj[i * NN + j] * rs[j];
    Lp[i * NPAD + j] = v;
    LpT[j * NPAD + i] = v;
  }
}

// ---------------------------------------------------------------------------
// Left contraction: D[e,n] = scale * sum_b LT[e,b] * S[b,n]  (- prev[e,n])
// LTp: padded 208x208.  S, D, prev: [200, 6400].
// One wave -> 16x64 output tile (4 accumulators share the A fragment).
// ---------------------------------------------------------------------------
__global__ void gemm_left(const float* __restrict__ LTp,
                          const float* __restrict__ S,
                          const float* __restrict__ prev,
                          float* __restrict__ D,
                          float scale) {
  int wid = (blockIdx.x * blockDim.x + threadIdx.x) >> 5;
  if (wid >= 13 * 100) return;           // wave-uniform exit (EXEC stays full)
  int mt = wid / 100, nt = wid % 100;
  int lane = threadIdx.x & 31;
  int l16 = lane & 15, half = lane >> 4;
  int mbase = mt * 16, nbase = nt * 64;

  v8f acc0 = {}, acc1 = {}, acc2 = {}, acc3 = {};
  const float* arow = LTp + (mbase + l16) * NPAD + 2 * half;
  const float* pb0  = S + (2 * half) * NCOL + nbase + l16;  // row k+2h
  const float* pb1  = pb0 + NCOL;                           // row k+2h+1

  for (int k = 0; k < NN; k += 4) {      // K=200 exact: no predication
    v2f a;
    a.x = arow[k];
    a.y = arow[k + 1];
    v2f b0, b1, b2, b3;
    b0.x = pb0[0];  b0.y = pb1[0];
    b1.x = pb0[16]; b1.y = pb1[16];
    b2.x = pb0[32]; b2.y = pb1[32];
    b3.x = pb0[48]; b3.y = pb1[48];
    pb0 += 4 * NCOL;
    pb1 += 4 * NCOL;
    acc0 = wmma4(a, b0, acc0);
    acc1 = wmma4(a, b1, acc1);
    acc2 = wmma4(a, b2, acc2);
    acc3 = wmma4(a, b3, acc3);
  }

#pragma unroll
  for (int g = 0; g < 8; ++g) {
    int m = mbase + g + half * 8;
    if (m < NN) {
      int col = nbase + l16;
      float* pd = D + m * NCOL + col;
      float v0 = scale * acc0[g];
      float v1 = scale * acc1[g];
      float v2 = scale * acc2[g];
      float v3 = scale * acc3[g];
      if (prev) {
        const float* pp = prev + m * NCOL + col;
        v0 -= pp[0]; v1 -= pp[16]; v2 -= pp[32]; v3 -= pp[48];
      }
      pd[0]  = v0;
      pd[16] = v1;
      pd[32] = v2;
      pd[48] = v3;
    }
  }
}

// ---------------------------------------------------------------------------
// Right contraction, batched over i:
//   D[i, e, d] = scale * sum_c Lp[e,c] * S[i, c, d]  (- prev[i, e, d])
// S, D, prev: [200, 200, 32] slices. One wave -> 16x32 tile of one i.
// ---------------------------------------------------------------------------
__global__ void gemm_right(const float* __restrict__ Lp,
                           const float* __restrict__ S,
                           const float* __restrict__ prev,
                           float* __restrict__ D,
                           float scale) {
  int mt = blockIdx.x * 8 + (threadIdx.x >> 5);
  if (mt >= 13) return;                  // wave-uniform exit
  int i = blockIdx.y;
  int lane = threadIdx.x & 31;
  int l16 = lane & 15, half = lane >> 4;
  int mbase = mt * 16;

  v8f acc0 = {}, acc1 = {};
  const float* arow = Lp + (mbase + l16) * NPAD + 2 * half;
  const float* pb0  = S + i * NCOL + (2 * half) * CCH + l16;
  const float* pb1  = pb0 + CCH;

  for (int k = 0; k < NN; k += 4) {      // K=200 exact: no predication
    v2f a;
    a.x = arow[k];
    a.y = arow[k + 1];
    v2f b0, b1;
    b0.x = pb0[0];  b0.y = pb1[0];
    b1.x = pb0[16]; b1.y = pb1[16];
    pb0 += 4 * CCH;
    pb1 += 4 * CCH;
    acc0 = wmma4(a, b0, acc0);
    acc1 = wmma4(a, b1, acc1);
  }

#pragma unroll
  for (int g = 0; g < 8; ++g) {
    int e = mbase + g + half * 8;
    if (e < NN) {
      float* pd = D + i * NCOL + e * CCH + l16;
      float v0 = scale * acc0[g];
      float v1 = scale * acc1[g];
      if (prev) {
        const float* pp = prev + i * NCOL + e * CCH + l16;
        v0 -= pp[0];
        v1 -= pp[16];
      }
      pd[0]  = v0;
      pd[16] = v1;
    }
  }
}

// ---------------------------------------------------------------------------
// Final projection for one batch:
//   out[r,o] = bias[o] + sum_{f<288} A[r,f] * W[o,f]
// where A[r, kk*32+c] = F[kk*SLICE + r*32 + c] (9 Chebyshev feature slices).
// M = 40000 (exactly 2500 tiles of 16), N = 64 (one 16x64 wave tile), K = 288.
// ---------------------------------------------------------------------------
__global__ void gemm_final(const float* __restrict__ F,
                           const float* __restrict__ W,     // [64, 288]
                           const float* __restrict__ bias,  // [64]
                           float* __restrict__ out) {        // [40000, 64]
  int wid = (blockIdx.x * blockDim.x + threadIdx.x) >> 5;
  if (wid >= 2500) return;               // wave-uniform exit
  int rbase = wid * 16;
  int lane = threadIdx.x & 31;
  int l16 = lane & 15, half = lane >> 4;

  v8f acc0 = {}, acc1 = {}, acc2 = {}, acc3 = {};

  const float* pw = W + l16 * 288 + 2 * half;           // advances by 4/iter
  const size_t rowoff = (size_t)(rbase + l16) * CCH + 2 * half;

  for (int kk = 0; kk < 9; ++kk) {       // feature slice (K block of 32)
    const float* pa = F + (size_t)kk * SLICE + rowoff;
#pragma unroll
    for (int c = 0; c < CCH; c += 4) {   // K within slice, exact
      v2f a;
      a.x = pa[c];
      a.y = pa[c + 1];
      v2f b0, b1, b2, b3;
      b0.x = pw[0];         b0.y = pw[1];
      b1.x = pw[16 * 288];  b1.y = pw[16 * 288 + 1];
      b2.x = pw[32 * 288];  b2.y = pw[32 * 288 + 1];
      b3.x = pw[48 * 288];  b3.y = pw[48 * 288 + 1];
      pw += 4;
      acc0 = wmma4(a, b0, acc0);
      acc1 = wmma4(a, b1, acc1);
      acc2 = wmma4(a, b2, acc2);
      acc3 = wmma4(a, b3, acc3);
    }
  }

  float bs0 = bias[l16];
  float bs1 = bias[16 + l16];
  float bs2 = bias[32 + l16];
  float bs3 = bias[48 + l16];

#pragma unroll
  for (int g = 0; g < 8; ++g) {
    int m = rbase + g + half * 8;        // always < 40000
    float* pd = out + (size_t)m * 64 + l16;
    pd[0]  = acc0[g] + bs0;
    pd[16] = acc1[g] + bs1;
    pd[32] = acc2[g] + bs2;
    pd[48] = acc3[g] + bs3;
  }
}

// ---------------------------------------------------------------------------
extern "C" void kernel_launch(void* const* d_in, const int* in_sizes, int n_in,
                              void* d_out, int out_size, void* d_ws, size_t ws_size,
                              hipStream_t stream) {
  const float* x      = (const float*)d_in[0];  // [8,200,200,32]
  const float* adj    = (const float*)d_in[1];  // [200,200]
  const float* weight = (const float*)d_in[2];  // [64,288]
  const float* bias   = (const float*)d_in[3];  // [64]
  float* out = (float*)d_out;                   // [8,200,200,64]

  float* Lp  = (float*)d_ws;                    // 208*208
  float* LpT = Lp + NPAD * NPAD;                // 208*208
  float* F   = LpT + NPAD * NPAD;               // 9 * SLICE floats (~44 MB total ws)

  lap_kernel<<<1, 256, 0, stream>>>(adj, Lp, LpT);

  for (int a = 0; a < 8; ++a) {
    const float* xa = x + (size_t)a * SLICE;
    float* outa = out + (size_t)a * NN * NN * 64;

    // F0 = T_left^0 = x_a
    hipMemcpyAsync(F, xa, (size_t)SLICE * sizeof(float),
                   hipMemcpyDeviceToDevice, stream);
    // F3 = T_left^1 = L^T @ x_a        (1300 waves)
    gemm_left<<<163, 256, 0, stream>>>(LpT, xa, nullptr, F + 3 * SLICE, 1.f);
    // F6 = T_left^2 = 2*(L^T @ F3) - x_a
    gemm_left<<<163, 256, 0, stream>>>(LpT, F + 3 * SLICE, xa, F + 6 * SLICE, 2.f);

    for (int k1 = 0; k1 < 3; ++k1) {
      float* Tk = F + (size_t)(3 * k1) * SLICE;
      // R1 = right(Tk)                  (2600 waves)
      gemm_right<<<dim3(2, NN), 256, 0, stream>>>(Lp, Tk, nullptr, Tk + SLICE, 1.f);
      // R2 = 2*right(R1) - Tk
      gemm_right<<<dim3(2, NN), 256, 0, stream>>>(Lp, Tk + SLICE, Tk, Tk + 2 * SLICE, 2.f);
    }

    // out_a = Fcat @ W^T + bias        (2500 waves)
    gemm_final<<<313, 256, 0, stream>>>(F, weight, bias, outa);
  }
}